// GNN_1520418423296
// MI455X (gfx1250) — compile-verified
//
#include <hip/hip_runtime.h>

typedef float v2f __attribute__((ext_vector_type(2)));
typedef float v8f __attribute__((ext_vector_type(8)));

#define NFEAT 128

static __device__ __forceinline__ unsigned hash32(unsigned x) {
  x ^= x >> 16; x *= 0x7feb352dU;
  x ^= x >> 15; x *= 0x846ca68bU;
  x ^= x >> 16;
  return x;
}

// ---------------------------------------------------------------------------
// deg / norm precompute
// ---------------------------------------------------------------------------
__global__ void k_deg_init(float* __restrict__ deg, int n) {
  int i = blockIdx.x * blockDim.x + threadIdx.x;
  if (i < n) deg[i] = 1.0f;  // self-loop weight
}

__global__ void k_deg_edges(float* __restrict__ deg,
                            const long long* __restrict__ col,
                            const float* __restrict__ ew, long long ne) {
  long long i = (long long)blockIdx.x * blockDim.x + threadIdx.x;
  if (i < ne) atomicAdd(&deg[col[i]], ew[i]);
}

__global__ void k_deg_fin(float* __restrict__ dis, int n) {
  int i = blockIdx.x * blockDim.x + threadIdx.x;
  if (i < n) {
    float d = dis[i];
    dis[i] = d > 0.0f ? rsqrtf(d) : 0.0f;
  }
}

__global__ void k_edge_norm(float* __restrict__ norm,
                            const float* __restrict__ dis,
                            const long long* __restrict__ row,
                            const long long* __restrict__ col,
                            const float* __restrict__ ew, long long ne) {
  long long i = (long long)blockIdx.x * blockDim.x + threadIdx.x;
  if (i < ne) norm[i] = dis[row[i]] * ew[i] * dis[col[i]];
}

// ---------------------------------------------------------------------------
// WMMA fp32 GEMM:  H[n x DOUT] = X[n x 128] @ W[128 x DOUT]
// One block = 16 rows; (DOUT/16) waves, each owning a 16-wide column tile.
// X tile staged in LDS (row stride 132 floats -> conflict-free strided reads).
// A-fragment layout (V_WMMA_F32_16X16X4_F32): lanes 0-15 hold K=0,1 for M=lane,
// lanes 16-31 hold K=2,3 for M=lane-16.  B mirrors with N=lane&15.
// D layout: VGPR v -> (M = v + 8*(lane>=16), N = lane&15).
// ---------------------------------------------------------------------------
template <int DOUT>
__global__ __launch_bounds__(256) void k_gemm_wmma(
    const float* __restrict__ X, const float* __restrict__ W,
    float* __restrict__ H, int nrows) {
  __shared__ float xs[16][NFEAT + 4];

  const int tid = threadIdx.x;
  const int lane = tid & 31;
  const int wave = tid >> 5;
  const int rowBase = blockIdx.x * 16;

  for (int idx = tid; idx < 16 * NFEAT; idx += blockDim.x) {
    int r = idx >> 7;
    int c = idx & (NFEAT - 1);
    int gr = rowBase + r;
    xs[r][c] = (gr < nrows) ? X[(size_t)gr * NFEAT + c] : 0.0f;
  }
  __syncthreads();

  const int m = lane & 15;
  const int koff = (lane >> 4) << 1;  // 0 or 2
  const int colBase = wave * 16;

  v8f acc = {};
#pragma unroll
  for (int k = 0; k < NFEAT; k += 4) {
    v2f a, b;
    a[0] = xs[m][k + koff];
    a[1] = xs[m][k + koff + 1];
    b[0] = W[(size_t)(k + koff) * DOUT + colBase + m];
    b[1] = W[(size_t)(k + koff + 1) * DOUT + colBase + m];
    acc = __builtin_amdgcn_wmma_f32_16x16x4_f32(
        /*neg_a=*/false, a, /*neg_b=*/false, b,
        /*c_mod=*/(short)0, acc, /*reuse_a=*/false, /*reuse_b=*/false);
  }

  const int mBase = (lane >> 4) << 3;  // 0 or 8
#pragma unroll
  for (int v = 0; v < 8; ++v) {
    int gr = rowBase + mBase + v;
    if (gr < nrows) H[(size_t)gr * DOUT + colBase + m] = acc[v];
  }
}

// ---------------------------------------------------------------------------
// Aggregation: out = D^-1/2 (A+I) D^-1/2 h + b
// ---------------------------------------------------------------------------
template <int DOUT>
__global__ void k_agg_init(float* __restrict__ out, const float* __restrict__ h,
                           const float* __restrict__ dis,
                           const float* __restrict__ bias, int n) {
  constexpr int CH = DOUT / 4;  // float4 chunks per row
  long long idx = (long long)blockIdx.x * blockDim.x + threadIdx.x;
  long long total = (long long)n * CH;
  if (idx >= total) return;
  int node = (int)(idx / CH);
  int ch = (int)(idx % CH);
  float s = dis[node];
  s = s * s;  // self-loop norm: dis*1*dis
  float4 hv = reinterpret_cast<const float4*>(h)[idx];
  float4 bv = reinterpret_cast<const float4*>(bias)[ch];
  float4 o;
  o.x = s * hv.x + bv.x;
  o.y = s * hv.y + bv.y;
  o.z = s * hv.z + bv.z;
  o.w = s * hv.w + bv.w;
  reinterpret_cast<float4*>(out)[idx] = o;
}

template <int DOUT>
__global__ void k_agg_edges(float* __restrict__ out,
                            const float* __restrict__ h,
                            const float* __restrict__ norm,
                            const long long* __restrict__ row,
                            const long long* __restrict__ col, long long ne) {
  constexpr int CH = DOUT / 4;   // float4 chunks per node row
  constexpr int EPW = 32 / CH;   // edges handled per wave
  const int lane = threadIdx.x & 31;
  const long long waveId =
      ((long long)blockIdx.x * blockDim.x + threadIdx.x) >> 5;
  const long long e = waveId * EPW + lane / CH;
  if (e >= ne) return;
  const int ch = lane % CH;
  const float w = norm[e];
  const long long r = row[e];
  const long long c = col[e];
  float4 hv = reinterpret_cast<const float4*>(h)[r * CH + ch];
  float* o = out + (size_t)c * DOUT + (size_t)ch * 4;
  atomicAdd(o + 0, w * hv.x);
  atomicAdd(o + 1, w * hv.y);
  atomicAdd(o + 2, w * hv.z);
  atomicAdd(o + 3, w * hv.w);
}

// relu + dropout(p=0.5) with deterministic hash RNG, scale = 1/(1-p) = 2
__global__ void k_relu_dropout(float* __restrict__ x, long long n,
                               unsigned seed) {
  long long i = (long long)blockIdx.x * blockDim.x + threadIdx.x;
  if (i >= n) return;
  float v = x[i];
  v = v > 0.0f ? v : 0.0f;
  unsigned hc = hash32(seed ^ (unsigned)i ^ ((unsigned)(i >> 32) * 0x9E3779B9u));
  x[i] = (hc & 1u) ? 2.0f * v : 0.0f;
}

// ---------------------------------------------------------------------------
static inline int cdiv_ll(long long a, long long b) {
  return (int)((a + b - 1) / b);
}

extern "C" void kernel_launch(void* const* d_in, const int* in_sizes, int n_in,
                              void* d_out, int out_size, void* d_ws,
                              size_t ws_size, hipStream_t stream) {
  const float* X = (const float*)d_in[0];
  const long long* ei = (const long long*)d_in[1];
  const float* ew = (const float*)d_in[2];
  const float* W1 = (const float*)d_in[3];
  const float* b1 = (const float*)d_in[4];
  const float* W2 = (const float*)d_in[5];
  const float* b2 = (const float*)d_in[6];
  const float* W3 = (const float*)d_in[7];
  const float* b3 = (const float*)d_in[8];
  float* out = (float*)d_out;

  const int n = in_sizes[0] / NFEAT;          // 100000
  const long long E = in_sizes[1] / 2;        // 3200000
  const long long* row = ei;
  const long long* col = ei + E;

  // workspace layout (floats)
  float* H = (float*)d_ws;                    // n*128
  float* XB = H + (size_t)n * NFEAT;          // n*128
  float* NORM = XB + (size_t)n * NFEAT;       // E
  float* DIS = NORM + E;                      // n

  const int T = 256;

  // --- degree / normalization (shared across all 3 layers) ---
  k_deg_init<<<cdiv_ll(n, T), T, 0, stream>>>(DIS, n);
  k_deg_edges<<<cdiv_ll(E, T), T, 0, stream>>>(DIS, col, ew, E);
  k_deg_fin<<<cdiv_ll(n, T), T, 0, stream>>>(DIS, n);
  k_edge_norm<<<cdiv_ll(E, T), T, 0, stream>>>(NORM, DIS, row, col, ew, E);

  const int gemmGrid = cdiv_ll(n, 16);
  const long long waves128 = (E + 0) / 1;       // 1 edge per wave (CH=32)
  const long long waves64 = (E + 1) / 2;        // 2 edges per wave (CH=16)

  // --- layer 1: 128 -> 128 ---
  k_gemm_wmma<128><<<gemmGrid, 256, 0, stream>>>(X, W1, H, n);
  k_agg_init<128><<<cdiv_ll((long long)n * 32, T), T, 0, stream>>>(XB, H, DIS,
                                                                   b1, n);
  k_agg_edges<128><<<cdiv_ll(waves128 * 32, T), T, 0, stream>>>(XB, H, NORM,
                                                                row, col, E);
  k_relu_dropout<<<cdiv_ll((long long)n * NFEAT, T), T, 0, stream>>>(
      XB, (long long)n * NFEAT, 0x1234567u);

  // --- layer 2: 128 -> 128 ---
  k_gemm_wmma<128><<<gemmGrid, 256, 0, stream>>>(XB, W2, H, n);
  k_agg_init<128><<<cdiv_ll((long long)n * 32, T), T, 0, stream>>>(XB, H, DIS,
                                                                   b2, n);
  k_agg_edges<128><<<cdiv_ll(waves128 * 32, T), T, 0, stream>>>(XB, H, NORM,
                                                                row, col, E);
  k_relu_dropout<<<cdiv_ll((long long)n * NFEAT, T), T, 0, stream>>>(
      XB, (long long)n * NFEAT, 0x89ABCDEFu);

  // --- layer 3: 128 -> 64 (no relu/dropout), writes d_out ---
  k_gemm_wmma<64><<<gemmGrid, 128, 0, stream>>>(XB, W3, H, n);
  k_agg_init<64><<<cdiv_ll((long long)n * 16, T), T, 0, stream>>>(out, H, DIS,
                                                                  b3, n);
  k_agg_edges<64><<<cdiv_ll(waves64 * 32, T), T, 0, stream>>>(out, H, NORM,
                                                              row, col, E);
}